// IWT_188978561035
// MI455X (gfx1250) — compile-verified
//
#include <hip/hip_runtime.h>

// Inverse Haar wavelet (IWT) for x:(32,256,64,64) f32 -> out:(32,64,128,128) f32.
//
// Butterfly expressed as D = A(16x4, constant Haar) x B(4x16, 16 pixels' subbands)
// via V_WMMA_F32_16X16X4_F32 (wave32).
//
// A layout (ISA 7.12.2, 32-bit A 16x4): lane m(0-15) holds A[m][0],A[m][1] in
// VGPR0,1; lane 16+m holds A[m][2],A[m][3]. Haar rows in A rows 0-3 AND 8-11
// (rows 4-7,12-15 zero) so that:
//   D vgpr0..3, lanes 0-15  : o00[n],o01[n] (M=0..3)
//   D vgpr2..3, lanes 16-31 : o10[n],o11[n] (M=10,11 = duplicated Haar rows 2,3)
// -> one b64 store per lane writes an interleaved (even,odd) output column pair,
//    lanes 0-15 -> output row 2h, lanes 16-31 -> output row 2h+1. Fully coalesced.
//
// B layout (mirror of A): lane n holds column n; VGPR0 = K rows {0|2}, VGPR1 =
// K rows {1|3} split across lane halves -> lanes 0-15 load cA,cH; lanes 16-31
// load cV,cD; each load = 2 contiguous 64B segments.

typedef __attribute__((ext_vector_type(2))) float v2f;
typedef __attribute__((ext_vector_type(8))) float v8f;

__global__ __launch_bounds__(256) void iwt_wmma_kernel(const float* __restrict__ x,
                                                       float* __restrict__ out) {
    const int tid  = blockIdx.x * blockDim.x + threadIdx.x;
    const int lane = threadIdx.x & 31;
    const int wid  = tid >> 5;          // one wave per (b,c,h) input row

    const int h  = wid & 63;            // H = 64
    const int bc = wid >> 6;            // b*64 + c, 0..2047

    const bool lo   = lane < 16;
    const int  m    = lane & 15;
    const int  comp = m & 3;            // output component: o00,o01,o10,o11
    const bool act  = (m & 4) == 0;     // A rows 0-3 and 8-11 carry the Haar matrix

    // Haar coefficients (x0.5), branchless so EXEC never diverges before WMMA.
    //   o00 = .5( A + H + V + D);  o01 = .5( A + H - V - D)
    //   o10 = .5( A - H + V - D);  o11 = .5( A - H - V + D)
    const float s0 = lo ? 0.5f
                        : (((comp & 1) == 0) ? 0.5f : -0.5f);            // k=0 (cA) | k=2 (cV)
    const float s1 = lo ? ((comp < 2) ? 0.5f : -0.5f)                    // k=1 (cH)
                        : ((comp == 0 || comp == 3) ? 0.5f : -0.5f);     // k=3 (cD)
    v2f a;
    a.x = act ? s0 : 0.0f;
    a.y = act ? s1 : 0.0f;

    // Input: x[b][c*4+sub][h][w] = bc*16384 + sub*4096 + h*64 + w
    // Output: out[b][c][r][s]    = bc*16384 + r*128 + s      (plane strides match!)
    const size_t plane = (size_t)bc * 16384;
    const int    sub0  = lo ? 0 : 2;    // B VGPR0: cA | cV
    const int    sub1  = lo ? 1 : 3;    // B VGPR1: cH | cD

    const float* in_b0 = x + plane + (size_t)sub0 * 4096 + (size_t)h * 64 + m;
    const float* in_b1 = x + plane + (size_t)sub1 * 4096 + (size_t)h * 64 + m;
    float* out_row = out + plane + (size_t)(2 * h + (lo ? 0 : 1)) * 128 + 2 * m;

#pragma unroll
    for (int g = 0; g < 4; ++g) {       // 4 groups of 16 pixels = one 64-pixel row
        v2f bm;
        bm.x = __builtin_nontemporal_load(in_b0 + g * 16);   // single-use stream: NT,
        bm.y = __builtin_nontemporal_load(in_b1 + g * 16);   // keep L2 for the output
        v8f cacc = {};
        // 8 args: (neg_a, A, neg_b, B, c_mod, C, reuse_a, reuse_b)
        v8f d = __builtin_amdgcn_wmma_f32_16x16x4_f32(
            false, a, false, bm, (short)0, cacc, false, false);

        float2 st;                       // lanes 0-15: (o00,o01); lanes 16-31: (o10,o11)
        st.x = lo ? d[0] : d[2];
        st.y = lo ? d[1] : d[3];
        *(float2*)(out_row + g * 32) = st;   // global_store_b64, 256B/wave, coalesced
    }
}

extern "C" void kernel_launch(void* const* d_in, const int* in_sizes, int n_in,
                              void* d_out, int out_size, void* d_ws, size_t ws_size,
                              hipStream_t stream) {
    const float* x = (const float*)d_in[0];
    float* out = (float*)d_out;

    // One wave consumes 256 input floats (64 pixels x 4 subbands).
    const int waves  = in_sizes[0] / 256;          // 131072 for (32,256,64,64)
    const int blocks = waves / 8;                  // 256 threads = 8 waves per block
    iwt_wmma_kernel<<<dim3(blocks), dim3(256), 0, stream>>>(x, out);
}